// DecoderStackWrapper_25890062861084
// MI455X (gfx1250) — compile-verified
//
#include <hip/hip_runtime.h>
#include <math.h>

#define NLAY 28
#define NH 16
#define NKV 8
#define HD 128
#define DMODEL 1024
#define FFD 3072
#define CACHE_LEN 2048
#define TTOT (CACHE_LEN + 1)
#define EPSV 1e-6f
#define ATT_SCALE 0.08838834764831845f /* 128^-0.5 */

typedef float v2f __attribute__((ext_vector_type(2)));
typedef float v8f __attribute__((ext_vector_type(8)));

// ---------------------------------------------------------------------------
// Async-copy the activation vector into LDS via the gfx1250 async-DMA path
// (GLOBAL_LOAD_ASYNC_TO_LDS_B128, tracked by ASYNCcnt).  LDS destination
// address = low 32 bits of the generic pointer (ISA: generic->LDS uses
// addr[31:0] as LDS_ADDR).  GV addressing: vaddr pair + saddr 'off'.
// ---------------------------------------------------------------------------
__device__ __forceinline__ void stage_x_async(float* xs, const float* __restrict__ x, int K) {
  for (int i = (int)threadIdx.x * 4; i < K; i += (int)blockDim.x * 4) {
    unsigned lds_addr = (unsigned)(unsigned long long)&xs[i];
    unsigned long long gaddr = (unsigned long long)(x + i);
    asm volatile("global_load_async_to_lds_b128 %0, %1, off"
                 :: "v"(lds_addr), "v"(gaddr) : "memory");
  }
  asm volatile("s_wait_asynccnt 0x0" ::: "memory");
  __syncthreads();
}

// ---------------------------------------------------------------------------
// One wave computes out[n0..n0+15] = xs[0..K) @ W with V_WMMA_F32_16X16X4_F32.
// Every lane carries the same A chunk (LDS broadcast) -> every WMMA row
// computes the same dot product; row M=0 (acc[0], lanes 0..15) is the result.
// No EXEC masking, no per-row selects.  B tiles stream straight from HBM with
// bumped pointers (read-once traffic = the bandwidth roofline term).
// ---------------------------------------------------------------------------
__device__ __forceinline__ float gemv16_wmma(const float* __restrict__ W,
                                             const float* __restrict__ xs,
                                             int K, int N, int n0) {
  const int    lane    = threadIdx.x & 31;
  const size_t strideN = (size_t)N;
  const int    koff    = (lane >= 16) ? 2 : 0;   // hi half-wave holds K=2,3
  const float* wp = W + (size_t)koff * strideN + (size_t)(n0 + (lane & 15));
  const float* xp = xs + koff;
  v8f acc = {0.f, 0.f, 0.f, 0.f, 0.f, 0.f, 0.f, 0.f};
#pragma unroll 4
  for (int k0 = 0; k0 < K; k0 += 4) {
    v2f a = *(const v2f*)xp;            // ds_load_b64 broadcast
    v2f b;
    b.x = wp[0];                        // B row ka   (64B coalesced / half-wave)
    b.y = wp[strideN];                  // B row ka+1
    __builtin_prefetch(wp + 16 * strideN, 0, 3);
    acc = __builtin_amdgcn_wmma_f32_16x16x4_f32(false, a, false, b,
                                                (short)0, acc, false, false);
    xp += 4;
    wp += 4 * strideN;
  }
  return acc[0];  // lanes 0..15 -> out[n0 + lane]
}

__device__ __forceinline__ float wave_sum(float p) {
  for (int off = 16; off; off >>= 1) p += __shfl_xor(p, off, 32);
  return p;
}
__device__ __forceinline__ float wave_max(float p) {
  for (int off = 16; off; off >>= 1) p = fmaxf(p, __shfl_xor(p, off, 32));
  return p;
}

// ---------------------------------------------------------------------------
// Old-cache relayout: (l,kv,2048,128) contiguous -> (l,kv,2049,128) strided.
// ---------------------------------------------------------------------------
__global__ void cache_copy_kernel(const float4* __restrict__ ksrc,
                                  const float4* __restrict__ vsrc,
                                  float4* __restrict__ kdst,
                                  float4* __restrict__ vdst) {
  const long long n4 = (long long)NLAY * NKV * CACHE_LEN * (HD / 4);
  const long long stride = (long long)gridDim.x * blockDim.x;
  for (long long i = (long long)blockIdx.x * blockDim.x + threadIdx.x; i < n4; i += stride) {
    const int d4       = (int)(i & (HD / 4 - 1));           // 0..31
    const int t        = (int)((i >> 5) & (CACHE_LEN - 1)); // 0..2047
    const long long lk = i >> 16;                           // 2048*32 = 2^16
    const long long dst = (lk * TTOT + t) * (HD / 4) + d4;
    kdst[dst] = ksrc[i];
    vdst[dst] = vsrc[i];
  }
}

__global__ void copy_kernel(const float* __restrict__ src, float* __restrict__ dst, int n) {
  int i = blockIdx.x * blockDim.x + threadIdx.x;
  if (i < n) dst[i] = src[i];
}

// ---------------------------------------------------------------------------
// RMSNorm over n elements (single block, 256 threads)
// ---------------------------------------------------------------------------
__global__ void rms_kernel(const float* __restrict__ x, const float* __restrict__ w,
                           float* __restrict__ out, int n) {
  __shared__ float red[8];
  float p = 0.f;
  for (int i = threadIdx.x; i < n; i += blockDim.x) { float v = x[i]; p += v * v; }
  p = wave_sum(p);
  if ((threadIdx.x & 31) == 0) red[threadIdx.x >> 5] = p;
  __syncthreads();
  if (threadIdx.x == 0) {
    float s = 0.f;
    for (int i = 0; i < (int)(blockDim.x >> 5); ++i) s += red[i];
    red[0] = rsqrtf(s / (float)n + EPSV);
  }
  __syncthreads();
  const float inv = red[0];
  for (int i = threadIdx.x; i < n; i += blockDim.x) out[i] = x[i] * inv * w[i];
}

// ---------------------------------------------------------------------------
// Fused QKV projection: 256 tiles (128 q / 64 k / 64 v), 8 waves per block.
// ---------------------------------------------------------------------------
__global__ void qkv_kernel(const float* __restrict__ hs,
                           const float* __restrict__ wq, const float* __restrict__ wk,
                           const float* __restrict__ wv,
                           float* __restrict__ qout, float* __restrict__ kout,
                           float* __restrict__ vout) {
  __shared__ __align__(16) float xs[DMODEL];
  stage_x_async(xs, hs, DMODEL);
  const int wave = threadIdx.x >> 5;
  const int tile = blockIdx.x * 8 + wave;  // 0..255
  const float* W; float* out; int N; int n0;
  if (tile < 128)      { W = wq; out = qout; N = NH * HD;  n0 = tile * 16; }
  else if (tile < 192) { W = wk; out = kout; N = NKV * HD; n0 = (tile - 128) * 16; }
  else                 { W = wv; out = vout; N = NKV * HD; n0 = (tile - 192) * 16; }
  const float r = gemv16_wmma(W, xs, DMODEL, N, n0);
  const int lane = threadIdx.x & 31;
  if (lane < 16) out[n0 + lane] = r;
}

// ---------------------------------------------------------------------------
// Per-head RMSNorm + RoPE.  blocks 0..15: Q heads -> qf.  16..23: K heads ->
// cache row 2048.  24..31: V heads -> cache row 2048 (plain copy).
// ---------------------------------------------------------------------------
__device__ __forceinline__ void norm_rope_store(const float* __restrict__ src,
                                                const float* __restrict__ wn,
                                                const float* __restrict__ cv,
                                                const float* __restrict__ sv,
                                                float* __restrict__ dst,
                                                float* sh, float* red) {
  const int tid = threadIdx.x;  // 128 threads
  const float v = src[tid];
  float p = wave_sum(v * v);
  if ((tid & 31) == 0) red[tid >> 5] = p;
  __syncthreads();
  const float inv = rsqrtf((red[0] + red[1] + red[2] + red[3]) / (float)HD + EPSV);
  const float nv = v * inv * wn[tid];
  sh[tid] = nv;
  __syncthreads();
  const float rot = (tid < HD / 2) ? -sh[tid + HD / 2] : sh[tid - HD / 2];
  dst[tid] = nv * cv[tid] + rot * sv[tid];
}

__global__ void head_finish_kernel(const float* __restrict__ qin,
                                   const float* __restrict__ kin,
                                   const float* __restrict__ vin,
                                   const float* __restrict__ qn, const float* __restrict__ kn,
                                   const float* __restrict__ cv, const float* __restrict__ sv,
                                   float* __restrict__ qf,
                                   float* __restrict__ kbase, float* __restrict__ vbase) {
  __shared__ float sh[HD];
  __shared__ float red[4];
  const int b = blockIdx.x;
  if (b < NH) {
    norm_rope_store(qin + b * HD, qn, cv, sv, qf + b * HD, sh, red);
  } else if (b < NH + NKV) {
    const int j = b - NH;
    norm_rope_store(kin + j * HD, kn, cv, sv,
                    kbase + ((size_t)j * TTOT + CACHE_LEN) * HD, sh, red);
  } else {
    const int j = b - NH - NKV;
    vbase[((size_t)j * TTOT + CACHE_LEN) * HD + threadIdx.x] = vin[j * HD + threadIdx.x];
  }
}

// ---------------------------------------------------------------------------
// Attention: one block per Q head (GQA group = head/2).  Wave-per-key score
// pass (coalesced 512B K rows), fp32 softmax, coalesced PV pass over dims.
// ---------------------------------------------------------------------------
__global__ void attn_kernel(const float* __restrict__ qf,
                            const float* __restrict__ kbase,
                            const float* __restrict__ vbase,
                            const float* __restrict__ mask,
                            float* __restrict__ attn_out) {
  __shared__ __align__(16) float q_s[HD];
  __shared__ float sc[TTOT];
  __shared__ float red[8];
  const int h  = blockIdx.x;
  const int kv = h >> 1;
  const float* __restrict__ Kc = kbase + (size_t)kv * TTOT * HD;
  const float* __restrict__ Vc = vbase + (size_t)kv * TTOT * HD;
  const int tid = threadIdx.x, wave = tid >> 5, lane = tid & 31;

  if (tid < HD) q_s[tid] = qf[h * HD + tid];
  __syncthreads();

  const float4* q4 = reinterpret_cast<const float4*>(q_s);
  const float4 qq = q4[lane];
  for (int t = wave; t < TTOT; t += 8) {
    const float4 kk = reinterpret_cast<const float4*>(Kc + (size_t)t * HD)[lane];
    float p = qq.x * kk.x + qq.y * kk.y + qq.z * kk.z + qq.w * kk.w;
    p = wave_sum(p);
    if (lane == 0) sc[t] = p * ATT_SCALE + mask[t];
  }
  __syncthreads();

  // max
  float m = -3.0e38f;
  for (int t = tid; t < TTOT; t += 256) m = fmaxf(m, sc[t]);
  m = wave_max(m);
  if (lane == 0) red[wave] = m;
  __syncthreads();
  if (tid == 0) {
    float mm = red[0];
    for (int i = 1; i < 8; ++i) mm = fmaxf(mm, red[i]);
    red[0] = mm;
  }
  __syncthreads();
  m = red[0];
  __syncthreads();

  // exp + sum
  float s = 0.f;
  for (int t = tid; t < TTOT; t += 256) { float e = expf(sc[t] - m); sc[t] = e; s += e; }
  s = wave_sum(s);
  if (lane == 0) red[wave] = s;
  __syncthreads();
  if (tid == 0) {
    float ss = 0.f;
    for (int i = 0; i < 8; ++i) ss += red[i];
    red[0] = ss;
  }
  __syncthreads();
  const float invs = 1.0f / red[0];
  __syncthreads();

  // P @ V, dims across threads (coalesced V rows)
  if (tid < HD) {
    float acc = 0.f;
    for (int t = 0; t < TTOT; ++t) acc += sc[t] * Vc[(size_t)t * HD + tid];
    attn_out[h * HD + tid] = acc * invs;
  }
}

// ---------------------------------------------------------------------------
// O projection + residual: 64 tiles (8 blocks x 8 waves), K = 2048.
// ---------------------------------------------------------------------------
__global__ void oproj_kernel(const float* __restrict__ x, const float* __restrict__ wo,
                             float* __restrict__ h) {
  __shared__ __align__(16) float xs[NH * HD];
  stage_x_async(xs, x, NH * HD);
  const int tile = blockIdx.x * 8 + (threadIdx.x >> 5);
  const int n0 = tile * 16;
  const float r = gemv16_wmma(wo, xs, NH * HD, DMODEL, n0);
  const int lane = threadIdx.x & 31;
  if (lane < 16) h[n0 + lane] += r;
}

// ---------------------------------------------------------------------------
// Fused gate/up + SiLU: 192 tiles (24 blocks x 8 waves), two WMMA chains.
// ---------------------------------------------------------------------------
__global__ void gateup_kernel(const float* __restrict__ hs2,
                              const float* __restrict__ wg, const float* __restrict__ wu,
                              float* __restrict__ act) {
  __shared__ __align__(16) float xs[DMODEL];
  stage_x_async(xs, hs2, DMODEL);
  const int tile = blockIdx.x * 8 + (threadIdx.x >> 5);
  const int n0 = tile * 16;
  const float g = gemv16_wmma(wg, xs, DMODEL, FFD, n0);
  const float u = gemv16_wmma(wu, xs, DMODEL, FFD, n0);
  const int lane = threadIdx.x & 31;
  if (lane < 16) {
    const float silu = g / (1.0f + expf(-g));
    act[n0 + lane] = silu * u;
  }
}

// ---------------------------------------------------------------------------
// Down projection + residual: 64 tiles (8 blocks x 8 waves), K = 3072.
// ---------------------------------------------------------------------------
__global__ void down_kernel(const float* __restrict__ act, const float* __restrict__ wd,
                            float* __restrict__ h) {
  __shared__ __align__(16) float xs[FFD];
  stage_x_async(xs, act, FFD);
  const int tile = blockIdx.x * 8 + (threadIdx.x >> 5);
  const int n0 = tile * 16;
  const float r = gemv16_wmma(wd, xs, FFD, DMODEL, n0);
  const int lane = threadIdx.x & 31;
  if (lane < 16) h[n0 + lane] += r;
}

// ---------------------------------------------------------------------------
extern "C" void kernel_launch(void* const* d_in, const int* in_sizes, int n_in,
                              void* d_out, int out_size, void* d_ws, size_t ws_size,
                              hipStream_t stream) {
  (void)in_sizes; (void)n_in; (void)out_size; (void)ws_size;
  const float* hidden = (const float*)d_in[0];
  const float* kc     = (const float*)d_in[1];
  const float* vc     = (const float*)d_in[2];
  const float* pcos   = (const float*)d_in[3];
  const float* psin   = (const float*)d_in[4];
  const float* mask   = (const float*)d_in[5];
  const float* ln1    = (const float*)d_in[6];
  const float* ln2    = (const float*)d_in[7];
  const float* qn     = (const float*)d_in[8];
  const float* kn     = (const float*)d_in[9];
  const float* wq     = (const float*)d_in[10];
  const float* wk     = (const float*)d_in[11];
  const float* wv     = (const float*)d_in[12];
  const float* wo     = (const float*)d_in[13];
  const float* wg     = (const float*)d_in[14];
  const float* wu     = (const float*)d_in[15];
  const float* wd     = (const float*)d_in[16];

  float* out = (float*)d_out;
  float* ws  = (float*)d_ws;
  float* h    = ws;            // 1024
  float* hs   = ws + 1024;     // 1024
  float* qbuf = ws + 2048;     // 2048
  float* kbuf = ws + 4096;     // 1024
  float* vbuf = ws + 5120;     // 1024
  float* qf   = ws + 6144;     // 2048
  float* attn = ws + 8192;     // 2048
  float* act  = ws + 10240;    // 3072
  float* hs2  = ws + 13312;    // 1024

  float* kout = out + 1024;
  float* vout = kout + (size_t)NLAY * NKV * TTOT * HD;

  cache_copy_kernel<<<8192, 256, 0, stream>>>((const float4*)kc, (const float4*)vc,
                                              (float4*)kout, (float4*)vout);
  copy_kernel<<<4, 256, 0, stream>>>(hidden, h, DMODEL);

  for (int l = 0; l < NLAY; ++l) {
    const float* wql = wq + (size_t)l * DMODEL * NH * HD;
    const float* wkl = wk + (size_t)l * DMODEL * NKV * HD;
    const float* wvl = wv + (size_t)l * DMODEL * NKV * HD;
    const float* wol = wo + (size_t)l * NH * HD * DMODEL;
    const float* wgl = wg + (size_t)l * DMODEL * FFD;
    const float* wul = wu + (size_t)l * DMODEL * FFD;
    const float* wdl = wd + (size_t)l * FFD * DMODEL;
    float* koutl = kout + (size_t)l * NKV * TTOT * HD;
    float* voutl = vout + (size_t)l * NKV * TTOT * HD;

    rms_kernel<<<1, 256, 0, stream>>>(h, ln1 + (size_t)l * DMODEL, hs, DMODEL);
    qkv_kernel<<<32, 256, 0, stream>>>(hs, wql, wkl, wvl, qbuf, kbuf, vbuf);
    head_finish_kernel<<<32, 128, 0, stream>>>(qbuf, kbuf, vbuf,
                                               qn + (size_t)l * HD, kn + (size_t)l * HD,
                                               pcos, psin, qf, koutl, voutl);
    attn_kernel<<<16, 256, 0, stream>>>(qf, koutl, voutl, mask, attn);
    oproj_kernel<<<8, 256, 0, stream>>>(attn, wol, h);
    rms_kernel<<<1, 256, 0, stream>>>(h, ln2 + (size_t)l * DMODEL, hs2, DMODEL);
    gateup_kernel<<<24, 256, 0, stream>>>(hs2, wgl, wul, act);
    down_kernel<<<8, 256, 0, stream>>>(act, wdl, h);
  }
  copy_kernel<<<4, 256, 0, stream>>>(h, out, DMODEL);
}